// GenODE_9242769622250
// MI455X (gfx1250) — compile-verified
//
#include <hip/hip_runtime.h>
#include <math.h>

// CDNA5 / gfx1250, wave32. fp32 WMMA 16x16x4 for the Linear(2->128) stage,
// native v_tanh_f32 for the activation, merging-butterfly for Linear(128->2).

typedef float v2f __attribute__((ext_vector_type(2)));
typedef float v8f __attribute__((ext_vector_type(8)));

#define HDIM   128
#define NTILES 8      // 128 / 16
#define STEPS  64     // fixed-step RK4; h^4 error ~6e-8 per step << 1e-5 tol

// Branchless tanh: native CDNA5 V_TANH_F32 (probe-confirmed to lower) or
// exp/rcp fallback.
__device__ __forceinline__ float fast_tanh(float x) {
#if __has_builtin(__builtin_amdgcn_tanhf)
  return __builtin_amdgcn_tanhf(x);
#else
  const float ax = __builtin_fabsf(x);
  const float e  = __expf(2.0f * ax);
  const float r  = __builtin_amdgcn_rcpf(e + 1.0f);
  const float t  = __builtin_fmaf(-2.0f, r, 1.0f);
  return __builtin_copysignf(t, x);
#endif
}

// Merging butterfly step: combines two lane-vectors a (destined for lanes with
// (lane&m)==0) and b (lanes with (lane&m)!=0) while folding lane-pairs {l,l^m}:
//   result(l) = a(l)+a(l^m)  if (l&m)==0
//             = b(l)+b(l^m)  otherwise
__device__ __forceinline__ float merge_red(float a, float b, int m, bool bit) {
  const float keep = bit ? b : a;
  const float send = bit ? a : b;
  return keep + __shfl_xor(send, m, 32);
}

struct VFParams {
  v2f   Bw1[NTILES];  // B tiles of W1^T (4x16: K=0 in .x / K=1 in .y on lanes<16; K=2,3 zero)
  float b1l[NTILES];  // b1[n*16 + lane%16] (C-matrix broadcast init)
  v2f   w2l[NTILES];  // sgn*W2[0][col], sgn*W2[1][col] at col = n*16 + lane%16
  float b2x, b2y;     // sgn*b2
};

// One batched RHS eval: f(y) = sgn * (W2 @ tanh(W1 y + b1) + b2)
// 16 states per wave (lanes 0-15 own state rows). All 32 lanes execute
// (EXEC must be all-ones for v_wmma); masking is done by zeroing A operands.
__device__ __forceinline__ void vf_eval(const VFParams& p, unsigned lane,
                                        float y0, float y1,
                                        float& f0, float& f1) {
  const bool lo = lane < 16u;
  // A-matrix 16x4 f32: VGPR0 = K0(lanes 0-15)|K2(16-31), VGPR1 = K1|K3. K=2,3 zero pad.
  v2f A;
  A.x = lo ? y0 : 0.0f;
  A.y = lo ? y1 : 0.0f;

  float pr0[8], pr1[8];   // partial dot of h-row (v) with W2 row 0 / row 1
#pragma unroll
  for (int v = 0; v < 8; ++v) { pr0[v] = 0.0f; pr1[v] = 0.0f; }

#pragma unroll
  for (int n = 0; n < NTILES; ++n) {
    v8f C;
#pragma unroll
    for (int v = 0; v < 8; ++v) C[v] = p.b1l[n];  // + b1 via accumulator init
    // D = A(16x4) x B(4x16) + C : 16x16 f32 tile of h_pre; rows v / v+8 per lane half
    v8f D = __builtin_amdgcn_wmma_f32_16x16x4_f32(
        false, A, false, p.Bw1[n], (short)0, C, false, false);
#pragma unroll
    for (int v = 0; v < 8; ++v) {
      float t = fast_tanh(D[v]);
      pr0[v] = fmaf(t, p.w2l[n].x, pr0[v]);
      pr1[v] = fmaf(t, p.w2l[n].y, pr1[v]);
    }
  }

  // Merging butterfly: collapse 8 row-quantities while reducing over lanes.
  // After masks 1,2,4 the register at lane l carries row (l&7) summed over its
  // 8-lane group; xor-8 completes the 16-lane sum.
  const bool b1_ = (lane & 1u) != 0u;
  const bool b2_ = (lane & 2u) != 0u;
  const bool b4_ = (lane & 4u) != 0u;

  float t0, t1, t2, t3, u0, u1, w0, w1;
  // d = 0
  t0 = merge_red(pr0[0], pr0[1], 1, b1_);
  t1 = merge_red(pr0[2], pr0[3], 1, b1_);
  t2 = merge_red(pr0[4], pr0[5], 1, b1_);
  t3 = merge_red(pr0[6], pr0[7], 1, b1_);
  u0 = merge_red(t0, t1, 2, b2_);
  u1 = merge_red(t2, t3, 2, b2_);
  w0 = merge_red(u0, u1, 4, b4_);
  w0 += __shfl_xor(w0, 8, 32);
  // d = 1
  t0 = merge_red(pr1[0], pr1[1], 1, b1_);
  t1 = merge_red(pr1[2], pr1[3], 1, b1_);
  t2 = merge_red(pr1[4], pr1[5], 1, b1_);
  t3 = merge_red(pr1[6], pr1[7], 1, b1_);
  u0 = merge_red(t0, t1, 2, b2_);
  u1 = merge_red(t2, t3, 2, b2_);
  w1 = merge_red(u0, u1, 4, b4_);
  w1 += __shfl_xor(w1, 8, 32);

  // Rows 8..15 live in the upper 16-lane half (same lane&7); swap halves and
  // select per lane: lane L (0..15) needs local value for rows 0..7, cross for 8..15.
  const float c0 = __shfl_xor(w0, 16, 32);
  const float c1 = __shfl_xor(w1, 16, 32);
  const bool  hi = (lane & 8u) != 0u;
  f0 = (hi ? c0 : w0) + p.b2x;    // sgn already folded into w2l / b2
  f1 = (hi ? c1 : w1) + p.b2y;
}

__global__ void __launch_bounds__(256)
genode_diag_rk4(const float* __restrict__ z_init,
                const float* __restrict__ W1, const float* __restrict__ b1,
                const float* __restrict__ W2, const float* __restrict__ b2,
                const int* __restrict__ dir,
                float* __restrict__ out, int N) {
  const unsigned tid  = blockIdx.x * blockDim.x + threadIdx.x;
  const unsigned lane = threadIdx.x & 31u;
  const unsigned wid  = tid >> 5;
  const unsigned l16  = lane & 15u;
  const bool     lo   = lane < 16u;

  const float sgn = (float)dir[0];

  // ---- preload the tiny MLP into per-lane registers (constant all solve) ----
  VFParams p;
#pragma unroll
  for (int n = 0; n < NTILES; ++n) {
    const int col = n * 16 + (int)l16;          // hidden-unit index 0..127
    const float w10 = W1[col * 2 + 0];          // W1 is [H,2] row-major
    const float w11 = W1[col * 2 + 1];
    p.Bw1[n].x = lo ? w10 : 0.0f;               // K=0 row of W1^T (K=2 half zero)
    p.Bw1[n].y = lo ? w11 : 0.0f;               // K=1 row        (K=3 half zero)
    p.b1l[n]   = b1[col];
    p.w2l[n].x = sgn * W2[col];                 // W2 is [2,H] row-major; sgn folded in
    p.w2l[n].y = sgn * W2[HDIM + col];
  }
  p.b2x = sgn * b2[0];
  p.b2y = sgn * b2[1];

  // ---- per-lane ODE state: lane L owns global state si = wid*16 + L ----
  const int  si     = (int)(wid * 16u + l16);
  const bool active = lo && (si < N);
  float z0 = active ? z_init[si * 2 + 0] : 0.0f;
  float z1 = active ? z_init[si * 2 + 1] : 0.0f;
  const float tEnd = (active && N > 1) ? ((float)si / (float)(N - 1)) : 0.0f;
  const float h    = tEnd * (1.0f / (float)STEPS);   // per-lane step (0 for idle lanes)

  // ---- classic RK4, 64 fixed steps over [0, t_i] ----
#pragma unroll 1
  for (int s = 0; s < STEPS; ++s) {
    float k10, k11, k20, k21, k30, k31, k40, k41;
    vf_eval(p, lane, z0, z1, k10, k11);
    vf_eval(p, lane, fmaf(0.5f * h, k10, z0), fmaf(0.5f * h, k11, z1), k20, k21);
    vf_eval(p, lane, fmaf(0.5f * h, k20, z0), fmaf(0.5f * h, k21, z1), k30, k31);
    vf_eval(p, lane, fmaf(h, k30, z0),        fmaf(h, k31, z1),        k40, k41);
    const float w = h * (1.0f / 6.0f);
    z0 += w * (k10 + 2.0f * k20 + 2.0f * k30 + k40);
    z1 += w * (k11 + 2.0f * k21 + 2.0f * k31 + k41);
  }

  if (active) {
    out[si * 2 + 0] = z0;
    out[si * 2 + 1] = z1;
  }
}

extern "C" void kernel_launch(void* const* d_in, const int* in_sizes, int n_in,
                              void* d_out, int out_size, void* d_ws, size_t ws_size,
                              hipStream_t stream) {
  (void)n_in; (void)out_size; (void)d_ws; (void)ws_size;
  const float* z_init = (const float*)d_in[0];
  const float* W1     = (const float*)d_in[1];
  const float* b1     = (const float*)d_in[2];
  const float* W2     = (const float*)d_in[3];
  const float* b2     = (const float*)d_in[4];
  const int*   dir    = (const int*)d_in[5];
  float* out = (float*)d_out;

  const int N       = in_sizes[0] / 2;               // number of states (D=2)
  const int waves   = (N + 15) / 16;                 // 16 states per wave32
  const int threads = waves * 32;
  const int block   = 256;                           // 8 waves per block
  const int grid    = (threads + block - 1) / block;

  genode_diag_rk4<<<grid, block, 0, stream>>>(z_init, W1, b1, W2, b2, dir, out, N);
}